// FadeevaEmulator_48034914238561
// MI455X (gfx1250) — compile-verified
//
#include <hip/hip_runtime.h>
#include <hip/hip_bf16.h>
#include <stdint.h>

#define SQRT2F      1.41421356237309504880f
#define SQRT2PIF    2.50662827463100050242f
#define PI_MATHF    3.14159265358979323846f
#define LSTRIDE     36          // floats per line record (9 header + 27 reciprocals; 144 B, 16B aligned)
#define MAX_LINES   128
#define WINDOW      25.0f       // |x| beyond this => contribution underflows to 0 in fp32

// workspace layout (floats):
//   [0 .. L*LSTRIDE)            : per-line records
//   [L*LSTRIDE .. L*LSTRIDE+2L) : compact filter table {lam, win_halfwidth} per line
//
// record: [0]=lam [1]=invs=1/(sqrt2*sigma) [2]=erfcx(y) [3]=1/(pi*y)
//         [4]=S1(y2) [5]=y/pi [6]=amp/(sqrt2pi*sigma) [7]=y [8]=win
//         [9..35] = 1/(an^2 + y^2), an = 0.5..13.5

// -------------------------------------------------------------------------
// Kernel 1: per-line precompute
// -------------------------------------------------------------------------
__global__ void voigt_line_precompute(const float* __restrict__ lam_centers,
                                      const float* __restrict__ log_amp,
                                      const float* __restrict__ log_sigma,
                                      const float* __restrict__ log_gamma,
                                      float* __restrict__ ws, int L)
{
    int l = blockIdx.x * blockDim.x + threadIdx.x;
    if (l >= L) return;

    float sigma = expf(log_sigma[l]);
    float gamma = expf(log_gamma[l]);
    float amp   = expf(log_amp[l]);

    float invs = 1.0f / (SQRT2F * sigma);
    float y    = gamma * invs;
    float y2   = y * y;
    float lam  = lam_centers[l];
    float win  = WINDOW / invs;                 // half-width in Angstroms of fp32-visible footprint

    // erfcx(y) = exp(y^2) * erfc(y); y > 0 and modest here, no overflow concern
    float erfcxy = expf(y2) * erfcf(y);

    // Sigma1 polynomial in y2 (the exx factor is applied per-point)
    float s1 = 0.778800786f    / (0.25f  + y2)
             + 0.36787945f     / (1.0f   + y2)
             + 0.105399221f    / (2.25f  + y2)
             + 0.0183156393f   / (4.0f   + y2)
             + 0.00193045416f  / (6.25f  + y2)
             + 0.000123409802f / (9.0f   + y2)
             + 4.78511765e-06f / (12.25f + y2)
             + 1.12535176e-07f / (16.0f  + y2);

    float* r = ws + (size_t)l * LSTRIDE;
    r[0] = lam;
    r[1] = invs;
    r[2] = erfcxy;
    r[3] = 1.0f / (PI_MATHF * y);
    r[4] = s1;
    r[5] = y * (1.0f / PI_MATHF);
    r[6] = amp / (SQRT2PIF * sigma);
    r[7] = y;
    r[8] = win;
#pragma unroll
    for (int n = 0; n < 27; ++n) {
        float an = 0.5f * (float)(n + 1);
        r[9 + n] = 1.0f / (an * an + y2);
    }

    // compact, coalesced filter table
    float* flt = ws + (size_t)L * LSTRIDE;
    flt[2 * l]     = lam;
    flt[2 * l + 1] = win;
}

// -------------------------------------------------------------------------
// Kernel 2: main spectrum kernel. One thread per wavelength (wl sorted).
// Per block: filter active lines, async-stage only their records to LDS.
// -------------------------------------------------------------------------
__global__ __launch_bounds__(256) void voigt_spectrum(
        const float* __restrict__ wl,
        const float* __restrict__ ws,
        const float* __restrict__ a_c,
        const float* __restrict__ b_c,
        const float* __restrict__ c_c,
        float* __restrict__ out, int W, int L)
{
    __shared__ float         tbl[MAX_LINES * LSTRIDE];  // compacted active-line records
    __shared__ int           act[MAX_LINES];
    __shared__ unsigned char pred[MAX_LINES];
    __shared__ int           nact_s;

    const int tid = threadIdx.x;
    const int b0  = blockIdx.x * blockDim.x;

    // block wavelength range (wl sorted ascending per reference setup)
    int hi_idx = b0 + blockDim.x - 1;
    if (hi_idx >= W) hi_idx = W - 1;
    const float wl_lo = wl[b0 < W ? b0 : (W - 1)];
    const float wl_hi = wl[hi_idx];

    // ---- 1. parallel line-window predicate (coalesced float2 reads) ----
    const float* flt = ws + (size_t)L * LSTRIDE;
    for (int l = tid; l < L; l += blockDim.x) {
        const float2 f = *(const float2*)(flt + 2 * l);   // {lam, win}
        pred[l] = (f.x >= wl_lo - f.y) && (f.x <= wl_hi + f.y);
    }
    __syncthreads();

    // ---- 2. deterministic in-order compaction (preserves fp product order) ----
    if (tid == 0) {
        int c = 0;
        for (int l = 0; l < L; ++l)
            if (pred[l]) act[c++] = l;
        nact_s = c;
    }
    __syncthreads();
    const int nact = nact_s;

    // ---- 3. async-stage only the active records: nact * 9 x 16B chunks ----
    {
        const unsigned lds_base = (unsigned)(uintptr_t)(void*)&tbl[0];
        const uint64_t gbase    = (uint64_t)(uintptr_t)ws;
        const int      nchunks  = nact * (LSTRIDE / 4);
        for (int c = tid; c < nchunks; c += blockDim.x) {
            const int line  = act[c / (LSTRIDE / 4)];
            const int chunk = c % (LSTRIDE / 4);
            unsigned ldsa = lds_base + (unsigned)c * 16u;                       // compact placement
            unsigned voff = (unsigned)line * (LSTRIDE * 4u) + (unsigned)chunk * 16u;
            asm volatile("global_load_async_to_lds_b128 %0, %1, %2"
                         :: "v"(ldsa), "v"(voff), "s"(gbase) : "memory");
        }
        asm volatile("s_wait_asynccnt 0x0" ::: "memory");
    }
    __syncthreads();

    // ---- 4. evaluate ----
    const int   w   = b0 + tid;
    const float wlv = (w < W) ? wl[w] : 0.0f;

    float prod = 1.0f;
    for (int i = 0; i < nact; ++i) {
        const float* r = &tbl[i * LSTRIDE];
        const float4 h0 = *(const float4*)r;            // lam, invs, erfcxy, 1/(pi*y)
        const float x = (wlv - h0.x) * h0.y;
        if (__builtin_fabsf(x) < WINDOW) {
            const float4 h1 = *(const float4*)(r + 4);  // s1, y/pi, pref, y
            const float y    = h1.w;
            const float xy   = x * y;
            const float exx  = __expf(-x * x);
            const float sxy  = __sinf(xy);
            const float c2xy = __cosf(2.0f * xy);
            // f = exx * ( erfcx(y)*cos(2xy) + sin^2(xy)/(pi*y) )
            const float f = exx * fmaf(h0.z, c2xy, sxy * sxy * h0.w);
            // 27-term Algorithm-916 sum
            float sig23 = 0.0f;
#pragma unroll
            for (int n = 0; n < 27; ++n) {
                const float an = 0.5f * (float)(n + 1);
                const float dp = an + x;
                const float dm = an - x;
                const float e  = __expf(-dp * dp) + __expf(-dm * dm);
                sig23 = fmaf(e, r[9 + n], sig23);
            }
            const float sig1 = exx * h1.x;
            const float re   = f + h1.y * fmaf(0.5f, sig23, -c2xy * sig1);
            prod *= fmaf(-h1.z, re, 1.0f);              // (1 - voigt)
        }
    }

    if (w < W) {
        const float ac = *a_c, bc = *b_c, cc = *c_c;
        const float wn = (wlv - 10500.0f) * (1.0f / 2500.0f);
        const float mod = fmaf(fmaf(cc, wn, bc), wn, ac);
        out[w] = prod * mod;
    }
}

// -------------------------------------------------------------------------
// Host launcher
// inputs: [0]=wl(W) [1]=lam_centers(L) [2]=log_amplitudes(L) [3]=log_sigma(L)
//         [4]=log_gamma(L) [5]=a_coeff [6]=b_coeff [7]=c_coeff
// -------------------------------------------------------------------------
extern "C" void kernel_launch(void* const* d_in, const int* in_sizes, int n_in,
                              void* d_out, int out_size, void* d_ws, size_t ws_size,
                              hipStream_t stream)
{
    const float* wl  = (const float*)d_in[0];
    const float* lam = (const float*)d_in[1];
    const float* lga = (const float*)d_in[2];
    const float* lgs = (const float*)d_in[3];
    const float* lgg = (const float*)d_in[4];
    const float* a_c = (const float*)d_in[5];
    const float* b_c = (const float*)d_in[6];
    const float* c_c = (const float*)d_in[7];
    float* out = (float*)d_out;

    const int W = in_sizes[0];
    int L = in_sizes[1];
    if (L > MAX_LINES) L = MAX_LINES;

    float* ws = (float*)d_ws;   // needs (L*LSTRIDE + 2L)*4 bytes (~15.2 KB for L=100)

    {
        dim3 blk(64), grd((unsigned)((L + 63) / 64));
        voigt_line_precompute<<<grd, blk, 0, stream>>>(lam, lga, lgs, lgg, ws, L);
    }
    {
        dim3 blk(256), grd((unsigned)((W + 255) / 256));
        voigt_spectrum<<<grd, blk, 0, stream>>>(wl, ws, a_c, b_c, c_c, out, W, L);
    }
}